// LiftBoostLearnWave_18348100288924
// MI455X (gfx1250) — compile-verified
//
#include <hip/hip_runtime.h>
#include <math.h>

// ---------------- problem constants ----------------
#define BB 64
#define CC 64
#define LL 4096
#define NLEV 3
#define NFILT 8
#define KK 4
#define EPSF 1e-5f

typedef __attribute__((ext_vector_type(2))) float v2f;
typedef __attribute__((ext_vector_type(8))) float v8f;

// ---------------- workspace layout (float offsets) ----------------
#define OFF_PING 0L                       // 8M floats (c-level0 / c-level2 / approx)
#define OFF_PONG (8L*1024*1024)           // 4M floats (c-level1)
#define OFF_LO   (12L*1024*1024)          // 8M floats (lo buffer; also trans-path d)
#define OFF_HI   (20L*1024*1024)          // 8M floats (hi buffer)
#define OFF_RS   (28L*1024*1024)          // 2*B*C = 8192 floats (per-row mu,var)
#define OFF_ACC  (28L*1024*1024 + 8192)   // 2048 floats of zeroed accumulators

// accumulator sub-offsets (within ACC)
#define A_BSTAT     0      // 12 slots * (2*BB)  = 1536  (branch-norm sum/sumsq per batch)
#define A_ENERGY    1536   // (lvl*2+s)*BB + b   = 384
#define A_SUMABS    1920   // 3  (sum |d| per level, details path)
#define A_SUMX      1923
#define A_SUMAPPROX 1924
#define A_ORTH      1925
#define A_EN        1926
#define A_EFFLO     1928   // 4 effective lo taps
#define A_EFFHI     1932   // 4 effective hi taps
#define ACC_N       2048

// ---------------- output layout (float offsets) ----------------
#define O_APPROX 0L
#define O_D0     2097152L
#define O_D1     10485760L
#define O_D2     14680064L
#define O_REGU   16777216L
#define O_ENERGY 16777217L
#define O_TRANS  16777601L

// ---------------- small device helpers ----------------
__device__ __forceinline__ int refl(int i, int n) {
  i = (i < 0) ? -i : i;
  i = (i >= n) ? (2 * n - 2 - i) : i;
  return i;
}

__device__ __forceinline__ float geluf(float x) {
  return 0.5f * x * (1.0f + erff(x * 0.70710678118654752f));
}

__device__ __forceinline__ float tap4(int t, float f0, float f1, float f2, float f3) {
  float v = 0.f;
  v = (t == 0) ? f0 : v;
  v = (t == 1) ? f1 : v;
  v = (t == 2) ? f2 : v;
  v = (t == 3) ? f3 : v;
  return v;
}

__device__ __forceinline__ float waveReduceSum(float v) {
#pragma unroll
  for (int off = 16; off > 0; off >>= 1) v += __shfl_xor(v, off, 32);
  return v;
}

__device__ __forceinline__ float blockReduceSum(float v, float* sm) {
  const int lane = threadIdx.x & 31, wid = threadIdx.x >> 5;
  v = waveReduceSum(v);
  __syncthreads();
  if (lane == 0) sm[wid] = v;
  __syncthreads();
  float r = 0.f;
  if (wid == 0) {
    r = (lane < (int)(blockDim.x >> 5)) ? sm[lane] : 0.f;
    r = waveReduceSum(r);
  }
  return r;  // valid in wave 0
}

// ---------------- WMMA conv tile ----------------
// D[m,n] = y[t0+16m+n],  y[t] = sum_k f[k]*zp[t+k],  zp = reflect-pad(z, leftpad, *)
// Banded factorization: D = sum_{i=0..4} A_i (16x4) * B_i (4x16),
//   A_i[m,k] = zp[t0+16m+4i+k],  B_i[k,n] = f[4i+k-n] if 0<=4i+k-n<=3 else 0.
// Wave32 layouts (ISA 7.12.2): A reg0/1 = K{0,1} (lanes 0-15) / K{2,3} (lanes 16-31);
// B assumed mirrored over K; D reg r holds rows r (lanes 0-15) and r+8 (lanes 16-31).
__device__ __forceinline__ v8f conv_tile(const float* z, int zstride, int lh,
                                         int leftpad, int t0,
                                         float f0, float f1, float f2, float f3) {
  const int lane = threadIdx.x & 31;
  const int mn = lane & 15;   // A row m == B col n per lane
  const int half = lane >> 4;
  v8f acc = {};
#pragma unroll
  for (int i = 0; i < 5; ++i) {
    const int j = t0 + 16 * mn + 4 * i + 2 * half;  // zp column for A reg0
    v2f a;
    a.x = z[(long)refl(j - leftpad, lh) * zstride];
    a.y = z[(long)refl(j + 1 - leftpad, lh) * zstride];
    const int tp = 4 * i + 2 * half - mn;           // filter tap for B reg0
    v2f b;
    b.x = tap4(tp, f0, f1, f2, f3);
    b.y = tap4(tp + 1, f0, f1, f2, f3);
    acc = __builtin_amdgcn_wmma_f32_16x16x4_f32(false, a, false, b, (short)0, acc,
                                                false, false);
  }
  return acc;
}

// ---------------- kernels ----------------
__global__ void k_zero(float* p, int n) {
  int i = blockIdx.x * blockDim.x + threadIdx.x;
  if (i < n) p[i] = 0.f;
}

// softmax(filter_weights) -> effective lo/hi taps; orth & energy-norm regularizers
__global__ void k_filters(const float* fw, const float* lof, const float* hif, float* acc) {
  if (threadIdx.x != 0 || blockIdx.x != 0) return;
  float m = -1e30f;
  for (int f = 0; f < NFILT; ++f) m = fmaxf(m, fw[f]);
  float w[NFILT]; float s = 0.f;
  for (int f = 0; f < NFILT; ++f) { w[f] = expf(fw[f] - m); s += w[f]; }
  const float inv = 1.f / s;
  for (int k = 0; k < KK; ++k) {
    float lo = 0.f, hi = 0.f;
    for (int f = 0; f < NFILT; ++f) {
      lo += w[f] * inv * lof[f * KK + k];
      hi += w[f] * inv * hif[f * KK + k];
    }
    acc[A_EFFLO + k] = lo;
    acc[A_EFFHI + k] = hi;
  }
  float orth = 0.f, en = 0.f;
  for (int f = 0; f < NFILT; ++f) {
    float dot = 0.f, nl = 0.f, nh = 0.f;
    for (int k = 0; k < KK; ++k) {
      float a = lof[f * KK + k], b = hif[f * KK + k];
      dot += a * b; nl += a * a; nh += b * b;
    }
    orth += fabsf(dot);
    en += fabsf(sqrtf(nl) - 1.f) + fabsf(sqrtf(nh) - 1.f);
  }
  acc[A_ORTH] = orth;
  acc[A_EN] = en;
}

// Shared-filter conv (details path lo/hi): reads ev/od view (stride 2) of cur,
// pad (1,2) reflect, stores raw result, accumulates per-batch sum|.| (energy).
__global__ void k_conv_lohi(const float* cur, long rowPitch, int zoff, int lh,
                            const float* eff, float* out, float* energyAcc,
                            int tilesPerRow) {
  const int wave = (blockIdx.x * blockDim.x + threadIdx.x) >> 5;
  const int tile = wave % tilesPerRow;
  const int row = wave / tilesPerRow;
  const float* z = cur + (long)row * rowPitch + zoff;
  const int t0 = tile * 256;
  if (t0 + 256 < lh) __builtin_prefetch(z + (long)(t0 + 256) * 2, 0, 1);
  v8f D = conv_tile(z, 2, lh, 1, t0, eff[0], eff[1], eff[2], eff[3]);
  const int lane = threadIdx.x & 31, n = lane & 15, half = lane >> 4;
  float s = 0.f;
#pragma unroll
  for (int r = 0; r < 8; ++r) {
    int t = t0 + 16 * (r + 8 * half) + n;
    out[(long)row * lh + t] = D[r];
    s += fabsf(D[r]);
  }
  s = waveReduceSum(s);
  if (lane == 0) atomicAdd(&energyAcc[row >> 6], s);  // b = row / CC
}

// Branch pass A: per-batch sum/sumsq of y = gelu(depthwise_conv_pad(2,1)(z)+bias)
__global__ void k_branch_stats(const float* z0, long rowPitch, int zoff, int zstride,
                               int lh, const float* W, const float* bias, float* bstat,
                               int tilesPerRow) {
  const int wave = (blockIdx.x * blockDim.x + threadIdx.x) >> 5;
  const int tile = wave % tilesPerRow;
  const int row = wave / tilesPerRow;
  const int c = row & (CC - 1), b = row >> 6;
  const float* z = z0 + (long)row * rowPitch + zoff;
  v8f D = conv_tile(z, zstride, lh, 2, tile * 256,
                    W[c * KK + 0], W[c * KK + 1], W[c * KK + 2], W[c * KK + 3]);
  const float bi = bias[c];
  float s = 0.f, s2 = 0.f;
#pragma unroll
  for (int r = 0; r < 8; ++r) {
    float y = geluf(D[r] + bi);
    s += y; s2 += y * y;
  }
  s = waveReduceSum(s);
  s2 = waveReduceSum(s2);
  if ((threadIdx.x & 31) == 0) {
    atomicAdd(&bstat[2 * b], s);
    atomicAdd(&bstat[2 * b + 1], s2);
  }
}

// Branch pass B: recompute y, normalize with batch stats, affine, combine with base.
__global__ void k_branch_apply(const float* z0, long rowPitch, int zoff, int zstride,
                               int lh, const float* W, const float* bias,
                               const float* g, const float* beta, const float* bstat,
                               const float* base0, long basePitch, int baseOff,
                               int baseStride, float sgn, float* out, float* absAcc,
                               int tilesPerRow) {
  const int wave = (blockIdx.x * blockDim.x + threadIdx.x) >> 5;
  const int tile = wave % tilesPerRow;
  const int row = wave / tilesPerRow;
  const int c = row & (CC - 1), b = row >> 6;
  const float invN = 1.f / (float)(CC * lh);
  const float mu = bstat[2 * b] * invN;
  const float var = bstat[2 * b + 1] * invN - mu * mu;
  const float isd = rsqrtf(var + EPSF);
  const float* z = z0 + (long)row * rowPitch + zoff;
  const int t0 = tile * 256;
  v8f D = conv_tile(z, zstride, lh, 2, t0,
                    W[c * KK + 0], W[c * KK + 1], W[c * KK + 2], W[c * KK + 3]);
  const float bi = bias[c];
  const int lane = threadIdx.x & 31, n = lane & 15, half = lane >> 4;
  const float* baser = base0 + (long)row * basePitch + baseOff;
  float s = 0.f;
#pragma unroll
  for (int r = 0; r < 8; ++r) {
    int t = t0 + 16 * (r + 8 * half) + n;
    float y = geluf(D[r] + bi);
    float v = (y - mu) * isd * g[(long)c * lh + t] + beta[(long)c * lh + t];
    float o = baser[(long)t * baseStride] + sgn * v;
    out[(long)row * lh + t] = o;
    s += fabsf(o);
  }
  if (absAcc) {
    s = waveReduceSum(s);
    if (lane == 0) atomicAdd(absAcc, s);
  }
}

// per-(b,c) mu/var over last axis
__global__ void k_rowstats(const float* d, int lh, float* rs) {
  const int row = blockIdx.x;
  const float* p = d + (long)row * lh;
  float s = 0.f, s2 = 0.f;
  for (int t = threadIdx.x; t < lh; t += blockDim.x) {
    float v = p[t];
    s += v; s2 += v * v;
  }
  __shared__ float sm[8];
  s = blockReduceSum(s, sm);
  s2 = blockReduceSum(s2, sm);
  if (threadIdx.x == 0) {
    float mu = s / lh;
    rs[2 * row] = mu;
    rs[2 * row + 1] = s2 / lh - mu * mu;
  }
}

// trans feature: mean_{c,t} ((d-mu_row)/max(std_row,1e-8))^4 - 3
__global__ void k_kurt(const float* d, int lh, const float* rs, float* transOut, int lvl) {
  const int b = blockIdx.x;
  float acc = 0.f;
  for (int idx = threadIdx.x; idx < CC * lh; idx += blockDim.x) {
    int c = idx / lh, t = idx - c * lh;
    int row = b * CC + c;
    float sig = fmaxf(sqrtf(rs[2 * row + 1]), 1e-8f);
    float z = (d[(long)row * lh + t] - rs[2 * row]) / sig;
    float z2 = z * z;
    acc += z2 * z2;
  }
  __shared__ float sm[8];
  acc = blockReduceSum(acc, sm);
  if (threadIdx.x == 0) transOut[b * NLEV + lvl] = acc / (float)(CC * lh) - 3.0f;
}

__global__ void k_inorm(const float* src, float* dst, int lh, const float* rs, long total) {
  long i = (long)blockIdx.x * blockDim.x + threadIdx.x;
  if (i >= total) return;
  long row = i / lh;
  dst[i] = (src[i] - rs[2 * row]) * rsqrtf(rs[2 * row + 1] + EPSF);
}

__global__ void k_sum(const float* p, long n, float* acc) {
  long i = (long)blockIdx.x * blockDim.x + threadIdx.x;
  const long stride = (long)gridDim.x * blockDim.x;
  float s = 0.f;
  for (; i < n; i += stride) s += p[i];
  __shared__ float sm[8];
  s = blockReduceSum(s, sm);
  if (threadIdx.x == 0) atomicAdd(acc, s);
}

__global__ void k_energy_fin(const float* eacc, float* out) {
  int i = blockIdx.x * blockDim.x + threadIdx.x;
  if (i >= BB * 2 * NLEV) return;
  int b = i / (2 * NLEV), col = i % (2 * NLEV);
  int lvl = col >> 1, s = col & 1;
  int lh = LL >> (lvl + 1);
  out[i] = eacc[(lvl * 2 + s) * BB + b] / (float)(CC * lh);
}

__global__ void k_regu(const float* acc, float* out) {
  if (threadIdx.x != 0 || blockIdx.x != 0) return;
  float r = 0.f;
  r += 0.1f * acc[A_SUMABS + 0] / (float)((long)BB * CC * 2048);
  r += 0.1f * acc[A_SUMABS + 1] / (float)((long)BB * CC * 1024);
  r += 0.1f * acc[A_SUMABS + 2] / (float)((long)BB * CC * 512);
  float xm = acc[A_SUMX] / (float)((long)BB * CC * LL);
  float am = acc[A_SUMAPPROX] / (float)((long)BB * CC * 512);
  r += 0.05f * fabsf(am - xm);
  r += 0.01f * acc[A_ORTH] + 0.01f * acc[A_EN];
  out[0] = r;
}

// ---------------- host orchestration ----------------
extern "C" void kernel_launch(void* const* d_in, const int* in_sizes, int n_in,
                              void* d_out, int out_size, void* d_ws, size_t ws_size,
                              hipStream_t stream) {
  (void)in_sizes; (void)n_in; (void)out_size; (void)ws_size;
  // Assumed flattening: x, filter_weights, hi_filters, lo_filters,
  // then per level: P_b, P_beta, P_g, P_w, U_b, U_beta, U_g, U_w (JAX leaf order).
  const float* x = (const float*)d_in[0];
  const float* fw = (const float*)d_in[1];
  const float* hif = (const float*)d_in[2];
  const float* lof = (const float*)d_in[3];
  const float *Pb[NLEV], *Pbeta[NLEV], *Pg[NLEV], *Pw[NLEV];
  const float *Ub[NLEV], *Ubeta[NLEV], *Ug[NLEV], *Uw[NLEV];
  for (int l = 0; l < NLEV; ++l) {
    int base = 4 + 8 * l;
    Pb[l] = (const float*)d_in[base + 0];
    Pbeta[l] = (const float*)d_in[base + 1];
    Pg[l] = (const float*)d_in[base + 2];
    Pw[l] = (const float*)d_in[base + 3];
    Ub[l] = (const float*)d_in[base + 4];
    Ubeta[l] = (const float*)d_in[base + 5];
    Ug[l] = (const float*)d_in[base + 6];
    Uw[l] = (const float*)d_in[base + 7];
  }
  float* ws = (float*)d_ws;
  float* ping = ws + OFF_PING;
  float* pong = ws + OFF_PONG;
  float* lobuf = ws + OFF_LO;
  float* hibuf = ws + OFF_HI;
  float* rs = ws + OFF_RS;
  float* acc = ws + OFF_ACC;
  float* out = (float*)d_out;

  k_zero<<<(ACC_N + 255) / 256, 256, 0, stream>>>(acc, ACC_N);
  k_filters<<<1, 32, 0, stream>>>(fw, lof, hif, acc);
  k_sum<<<2048, 256, 0, stream>>>(x, (long)BB * CC * LL, acc + A_SUMX);

  float* cbufs[NLEV] = {ping, pong, ping};
  int slot = 0;

  // ---------------- trans-feature path ----------------
  {
    const float* cur = x;
    for (int lvl = 0; lvl < NLEV; ++lvl) {
      int lh = LL >> (lvl + 1);
      long Lc = 2L * lh;
      int tiles = lh / 256;
      int blocks = (BB * CC * tiles) / 8;  // 8 waves per 256-thread block
      float* cb = cbufs[lvl];
      float* bsU = acc + A_BSTAT + (slot++) * 2 * BB;
      k_branch_stats<<<blocks, 256, 0, stream>>>(cur, Lc, 1, 2, lh, Uw[lvl], Ub[lvl],
                                                 bsU, tiles);
      k_branch_apply<<<blocks, 256, 0, stream>>>(cur, Lc, 1, 2, lh, Uw[lvl], Ub[lvl],
                                                 Ug[lvl], Ubeta[lvl], bsU, cur, Lc, 0, 2,
                                                 +1.f, cb, nullptr, tiles);
      float* bsP = acc + A_BSTAT + (slot++) * 2 * BB;
      k_branch_stats<<<blocks, 256, 0, stream>>>(cb, lh, 0, 1, lh, Pw[lvl], Pb[lvl],
                                                 bsP, tiles);
      k_branch_apply<<<blocks, 256, 0, stream>>>(cb, lh, 0, 1, lh, Pw[lvl], Pb[lvl],
                                                 Pg[lvl], Pbeta[lvl], bsP, cur, Lc, 1, 2,
                                                 -1.f, lobuf, nullptr, tiles);
      k_rowstats<<<BB * CC, 256, 0, stream>>>(lobuf, lh, rs);
      k_kurt<<<BB, 256, 0, stream>>>(lobuf, lh, rs, out + O_TRANS, lvl);
      cur = cb;
    }
  }

  // ---------------- details path ----------------
  {
    const float* cur = x;
    float* dout[NLEV] = {out + O_D0, out + O_D1, out + O_D2};
    for (int lvl = 0; lvl < NLEV; ++lvl) {
      int lh = LL >> (lvl + 1);
      long Lc = 2L * lh;
      int tiles = lh / 256;
      int blocks = (BB * CC * tiles) / 8;
      float* cb = cbufs[lvl];
      k_conv_lohi<<<blocks, 256, 0, stream>>>(cur, Lc, 0, lh, acc + A_EFFLO, lobuf,
                                              acc + A_ENERGY + (lvl * 2 + 0) * BB, tiles);
      k_conv_lohi<<<blocks, 256, 0, stream>>>(cur, Lc, 1, lh, acc + A_EFFHI, hibuf,
                                              acc + A_ENERGY + (lvl * 2 + 1) * BB, tiles);
      float* bsU = acc + A_BSTAT + (slot++) * 2 * BB;
      k_branch_stats<<<blocks, 256, 0, stream>>>(hibuf, lh, 0, 1, lh, Uw[lvl], Ub[lvl],
                                                 bsU, tiles);
      k_branch_apply<<<blocks, 256, 0, stream>>>(hibuf, lh, 0, 1, lh, Uw[lvl], Ub[lvl],
                                                 Ug[lvl], Ubeta[lvl], bsU, lobuf, lh, 0, 1,
                                                 +1.f, cb, nullptr, tiles);
      float* bsP = acc + A_BSTAT + (slot++) * 2 * BB;
      k_branch_stats<<<blocks, 256, 0, stream>>>(cb, lh, 0, 1, lh, Pw[lvl], Pb[lvl],
                                                 bsP, tiles);
      k_branch_apply<<<blocks, 256, 0, stream>>>(cb, lh, 0, 1, lh, Pw[lvl], Pb[lvl],
                                                 Pg[lvl], Pbeta[lvl], bsP, hibuf, lh, 0, 1,
                                                 -1.f, dout[lvl], acc + A_SUMABS + lvl,
                                                 tiles);
      k_rowstats<<<BB * CC, 256, 0, stream>>>(dout[lvl], lh, rs);
      long total = (long)BB * CC * lh;
      k_inorm<<<(int)((total + 255) / 256), 256, 0, stream>>>(dout[lvl], dout[lvl], lh, rs,
                                                              total);
      cur = cb;
    }
    // approx = cbufs[2] (length 512 per row)
    float* approx = cbufs[2];
    k_sum<<<1024, 256, 0, stream>>>(approx, (long)BB * CC * 512, acc + A_SUMAPPROX);
    k_rowstats<<<BB * CC, 256, 0, stream>>>(approx, 512, rs);
    long tot = (long)BB * CC * 512;
    k_inorm<<<(int)((tot + 255) / 256), 256, 0, stream>>>(approx, out + O_APPROX, 512, rs,
                                                          tot);
  }

  k_energy_fin<<<2, 256, 0, stream>>>(acc + A_ENERGY, out + O_ENERGY);
  k_regu<<<1, 1, 0, stream>>>(acc, out + O_REGU);
}